// RefractorySpikeFunction_50826642981386
// MI455X (gfx1250) — compile-verified
//
#include <hip/hip_runtime.h>
#include <stddef.h>

// Problem dims (fixed by reference setup_inputs()).
#define B_    32
#define T_    4096
#define N_    256
#define RP_   5      // REFRACTORY_PERIOD

// Chunked-scan decomposition.
#define C_       32                // chunks along T
#define TC_      (T_ / C_)         // 128 timesteps per chunk
#define ST_      16                // timesteps per LDS subtile (16 KiB)
#define NSUB_    (TC_ / ST_)       // 8 subtiles per chunk
#define THREADS_ 256               // = N_, 8 wave32s per block

// ---- CDNA5 async global->LDS path (guarded: compiles either way) ----
#if defined(__has_builtin)
#if __has_builtin(__builtin_amdgcn_global_load_async_to_lds_b128)
#define USE_ASYNC 1
#endif
#endif

__device__ __forceinline__ void wait_async_all() {
#if defined(__has_builtin) && __has_builtin(__builtin_amdgcn_s_wait_asynccnt)
  __builtin_amdgcn_s_wait_asynccnt(0);
#else
  asm volatile("s_wait_asynccnt 0" ::: "memory");
#endif
}

#ifdef USE_ASYNC
// Builtin signature (from hipcc diagnostic): first param is
// 'int __attribute__((vector_size(4*sizeof(int)))) *' — typed int4 pointers,
// global (AS1) source and LDS (AS3) destination, plus imm offset and cpol.
typedef int v4i_ __attribute__((vector_size(4 * sizeof(int))));
typedef __attribute__((address_space(1))) v4i_* gv4i_p;
typedef __attribute__((address_space(3))) v4i_* lv4i_p;

// Copy one subtile (ST_ rows x N_ floats = 16 KiB) global -> LDS asynchronously.
// 256 threads x 4 x b128 (16 B each) = 16 KiB; fully coalesced, 16-B aligned.
__device__ __forceinline__ void issue_subtile(const float* gsrc, float* ldst, int tid) {
#pragma unroll
  for (int r = 0; r < (ST_ * N_ * 4) / (THREADS_ * 16); ++r) {  // 4 rounds
    const int off = r * THREADS_ * 16 + tid * 16;
    __builtin_amdgcn_global_load_async_to_lds_b128(
        (gv4i_p)((char*)gsrc + off),
        (lv4i_p)((char*)ldst + off),
        /*offset=*/0, /*cpol=*/0);
  }
}
#endif

// FSM step shared by all kernels (exactly mirrors the reference scan step).
__device__ __forceinline__ int fsm_step(int q, bool xpos, int* spike_out) {
  const int spike = (xpos && (q == 0)) ? 1 : 0;
  *spike_out = spike;
  return (q > 0 ? q - 1 : 0) + RP_ * spike;
}

// ---------------- Phase 1: per-chunk 6-state transition maps ----------------
__global__ __launch_bounds__(THREADS_) void refrac_phase1(
    const float* __restrict__ x, unsigned* __restrict__ maps) {
  const int blk = blockIdx.x;          // b * C_ + c
  const int b = blk / C_, c = blk % C_;
  const int tid = threadIdx.x;         // n
  const float* base = x + ((size_t)b * T_ + (size_t)c * TC_) * N_;

  int q[6];
#pragma unroll
  for (int s = 0; s < 6; ++s) q[s] = s;

#ifdef USE_ASYNC
  __shared__ __align__(16) float tile[2][ST_ * N_];
  issue_subtile(base, tile[0], tid);
  for (int s = 0; s < NSUB_; ++s) {
    wait_async_all();        // my subtile-s DMAs landed
    __syncthreads();         // everyone's landed; prev compute done -> WAR safe
    if (s + 1 < NSUB_)
      issue_subtile(base + (size_t)(s + 1) * ST_ * N_, tile[(s + 1) & 1], tid);
    const float* tl = tile[s & 1];
#pragma unroll
    for (int t = 0; t < ST_; ++t) {
      const bool xpos = tl[t * N_ + tid] > 0.0f;
#pragma unroll
      for (int k = 0; k < 6; ++k) {
        int sp;
        q[k] = fsm_step(q[k], xpos, &sp);
      }
    }
  }
#else
#pragma unroll 4
  for (int t = 0; t < TC_; ++t) {
    const bool xpos = base[(size_t)t * N_ + tid] > 0.0f;
#pragma unroll
    for (int k = 0; k < 6; ++k) {
      int sp;
      q[k] = fsm_step(q[k], xpos, &sp);
    }
  }
#endif

  unsigned w = 0;
#pragma unroll
  for (int k = 0; k < 6; ++k) w |= (unsigned)q[k] << (3 * k);
  maps[((size_t)b * C_ + c) * N_ + tid] = w;
}

// ---------------- Phase 2: compose maps -> entry state per chunk ------------
__global__ __launch_bounds__(THREADS_) void refrac_phase2(
    const unsigned* __restrict__ maps, unsigned* __restrict__ q0arr) {
  const int idx = blockIdx.x * THREADS_ + threadIdx.x;  // b*N_ + n
  const int b = idx / N_, n = idx % N_;
  unsigned q = 0;  // reference starts the scan at q0 = 0
  for (int c = 0; c < C_; ++c) {
    const size_t o = ((size_t)b * C_ + c) * N_ + n;
    q0arr[o] = q;
    q = (maps[o] >> (3 * q)) & 7u;
  }
}

// ---------------- Phase 3: replay chunks with known entry state, emit z -----
__global__ __launch_bounds__(THREADS_) void refrac_phase3(
    const float* __restrict__ x, const unsigned* __restrict__ q0arr,
    float* __restrict__ z) {
  const int blk = blockIdx.x;
  const int b = blk / C_, c = blk % C_;
  const int tid = threadIdx.x;
  const size_t tbase = ((size_t)b * T_ + (size_t)c * TC_) * N_;
  const float* base = x + tbase;
  float* zb = z + tbase;
  int q = (int)q0arr[((size_t)b * C_ + c) * N_ + tid];

#ifdef USE_ASYNC
  __shared__ __align__(16) float tile[2][ST_ * N_];
  issue_subtile(base, tile[0], tid);
  for (int s = 0; s < NSUB_; ++s) {
    wait_async_all();
    __syncthreads();
    if (s + 1 < NSUB_)
      issue_subtile(base + (size_t)(s + 1) * ST_ * N_, tile[(s + 1) & 1], tid);
    const float* tl = tile[s & 1];
    float* zo = zb + (size_t)s * ST_ * N_;
#pragma unroll
    for (int t = 0; t < ST_; ++t) {
      int sp;
      q = fsm_step(q, tl[t * N_ + tid] > 0.0f, &sp);
      zo[t * N_ + tid] = sp ? 1.0f : 0.0f;
    }
  }
#else
#pragma unroll 4
  for (int t = 0; t < TC_; ++t) {
    int sp;
    q = fsm_step(q, base[(size_t)t * N_ + tid] > 0.0f, &sp);
    zb[(size_t)t * N_ + tid] = sp ? 1.0f : 0.0f;
  }
#endif
}

// ---------------- Fallback: naive single-pass (if workspace too small) ------
__global__ __launch_bounds__(THREADS_) void refrac_naive(
    const float* __restrict__ x, float* __restrict__ z) {
  const int idx = blockIdx.x * THREADS_ + threadIdx.x;  // b*N_ + n
  const int b = idx / N_, n = idx % N_;
  const float* xb = x + (size_t)b * T_ * N_ + n;
  float* zb = z + (size_t)b * T_ * N_ + n;
  int q = 0;
  for (int t = 0; t < T_; ++t) {
    int sp;
    q = fsm_step(q, xb[(size_t)t * N_] > 0.0f, &sp);
    zb[(size_t)t * N_] = sp ? 1.0f : 0.0f;
  }
}

extern "C" void kernel_launch(void* const* d_in, const int* in_sizes, int n_in,
                              void* d_out, int out_size, void* d_ws, size_t ws_size,
                              hipStream_t stream) {
  (void)in_sizes; (void)n_in; (void)out_size;
  const float* x = (const float*)d_in[0];
  float* z = (float*)d_out;

  const size_t map_elems = (size_t)B_ * C_ * N_;             // 256K u32 = 1 MiB
  const size_t need = 2 * map_elems * sizeof(unsigned);      // maps + q0arr

  if (ws_size >= need) {
    unsigned* maps = (unsigned*)d_ws;
    unsigned* q0arr = maps + map_elems;
    refrac_phase1<<<B_ * C_, THREADS_, 0, stream>>>(x, maps);
    refrac_phase2<<<(B_ * N_) / THREADS_, THREADS_, 0, stream>>>(maps, q0arr);
    refrac_phase3<<<B_ * C_, THREADS_, 0, stream>>>(x, q0arr, z);
  } else {
    refrac_naive<<<(B_ * N_) / THREADS_, THREADS_, 0, stream>>>(x, z);
  }
}